// Mesh_loss_56796647522838
// MI455X (gfx1250) — compile-verified
//
#include <hip/hip_runtime.h>
#include <math.h>

typedef __attribute__((ext_vector_type(2))) float v2f;
typedef __attribute__((ext_vector_type(8))) float v8f;

#define THREADS 256

// ------------------------------------------------------------------
// small device helpers
// ------------------------------------------------------------------
__device__ inline float dot3p(const float* a, const float* b) {
  return a[0]*b[0] + a[1]*b[1] + a[2]*b[2];
}
__device__ inline float dist2p(const float* a, const float* b) {
  float dx = a[0]-b[0], dy = a[1]-b[1], dz = a[2]-b[2];
  return dx*dx + dy*dy + dz*dz;
}
// cotangent weights per face corner (matches reference _cot_edges)
__device__ inline void cot_weights(const float* p0, const float* p1, const float* p2, float* cw) {
  float A2 = dist2p(p1, p2);
  float B2 = dist2p(p0, p2);
  float C2 = dist2p(p0, p1);
  float Al = sqrtf(A2), Bl = sqrtf(B2), Cl = sqrtf(C2);
  float s  = 0.5f * (Al + Bl + Cl);
  float t  = s * (s - Al) * (s - Bl) * (s - Cl);
  float ar = sqrtf(fmaxf(t, 1e-12f));
  float inv = 1.0f / (4.0f * ar);
  cw[0] = (B2 + C2 - A2) * inv;
  cw[1] = (A2 + C2 - B2) * inv;
  cw[2] = (A2 + B2 - C2) * inv;
}

// ------------------------------------------------------------------
// zero fill
// ------------------------------------------------------------------
__global__ void zero_kernel(float* __restrict__ p, int n) {
  int i = blockIdx.x * blockDim.x + threadIdx.x;
  if (i < n) p[i] = 0.0f;
}

// ------------------------------------------------------------------
// face areas (per batch, uses deformed verts)
// ------------------------------------------------------------------
__global__ void face_area_kernel(const float* __restrict__ verts, const int* __restrict__ faces,
                                 float* __restrict__ area, int F, int Vn, int nB) {
  int gid = blockIdx.x * blockDim.x + threadIdx.x;
  if (gid >= nB * F) return;
  int b = gid / F, f = gid - b * F;
  const float* vb = verts + (size_t)b * Vn * 3;
  const float* p0 = vb + 3 * faces[3*f + 0];
  const float* p1 = vb + 3 * faces[3*f + 1];
  const float* p2 = vb + 3 * faces[3*f + 2];
  float e1x = p1[0]-p0[0], e1y = p1[1]-p0[1], e1z = p1[2]-p0[2];
  float e2x = p2[0]-p0[0], e2y = p2[1]-p0[1], e2z = p2[2]-p0[2];
  float cx = e1y*e2z - e1z*e2y;
  float cy = e1z*e2x - e1x*e2z;
  float cz = e1x*e2y - e1y*e2x;
  area[gid] = 0.5f * sqrtf(cx*cx + cy*cy + cz*cz);
}

// ------------------------------------------------------------------
// inclusive scan: per-block scan + block partials, partial scan, add-back
// ------------------------------------------------------------------
__global__ void scan_block_kernel(const float* __restrict__ in, float* __restrict__ out,
                                  float* __restrict__ bsum, int n, int nblk) {
  __shared__ float sh[256];
  int b = blockIdx.y;
  int tid = threadIdx.x;
  int i = blockIdx.x * 256 + tid;
  float v = (i < n) ? in[(size_t)b * n + i] : 0.0f;
  sh[tid] = v;
  __syncthreads();
  for (int off = 1; off < 256; off <<= 1) {
    float t = (tid >= off) ? sh[tid - off] : 0.0f;
    __syncthreads();
    sh[tid] += t;
    __syncthreads();
  }
  if (i < n) out[(size_t)b * n + i] = sh[tid];
  if (tid == 255) bsum[(size_t)b * nblk + blockIdx.x] = sh[255];
}

__global__ void scan_partials_kernel(float* __restrict__ bsum, int nblk, int nB) {
  int b = blockIdx.x * blockDim.x + threadIdx.x;
  if (b >= nB) return;
  float acc = 0.0f;
  for (int i = 0; i < nblk; ++i) {
    acc += bsum[(size_t)b * nblk + i];
    bsum[(size_t)b * nblk + i] = acc;
  }
}

__global__ void scan_add_kernel(float* __restrict__ pref, const float* __restrict__ bsum,
                                int n, int nblk) {
  int b = blockIdx.y;
  int i = blockIdx.x * 256 + threadIdx.x;
  if (i < n && blockIdx.x > 0)
    pref[(size_t)b * n + i] += bsum[(size_t)b * nblk + blockIdx.x - 1];
}

// ------------------------------------------------------------------
// area-weighted surface sampling (searchsorted + barycentric)
// ------------------------------------------------------------------
__global__ void sample_kernel(const float* __restrict__ verts_scr, const int* __restrict__ faces,
                              const float* __restrict__ prefix, const float* __restrict__ u_face,
                              const float* __restrict__ r_bary, float* __restrict__ smp,
                              float* __restrict__ sqs, int F, int Vn, int S, int nB) {
  int gid = blockIdx.x * blockDim.x + threadIdx.x;
  if (gid >= nB * S) return;
  int b = gid / S;
  const float* pf = prefix + (size_t)b * F;
  float total = pf[F - 1];
  float key = u_face[gid] * total;
  int lo = 0, hi = F - 1;
  while (lo < hi) {
    int mid = (lo + hi) >> 1;
    if (pf[mid] >= key) hi = mid; else lo = mid + 1;
  }
  const float* vb = verts_scr + (size_t)b * Vn * 3;
  const float* p0 = vb + 3 * faces[3*lo + 0];
  const float* p1 = vb + 3 * faces[3*lo + 1];
  const float* p2 = vb + 3 * faces[3*lo + 2];
  float r0 = r_bary[2*gid + 0], r1 = r_bary[2*gid + 1];
  float su = sqrtf(r0);
  float w0 = 1.0f - su, w1 = su * (1.0f - r1), w2 = su * r1;
  float px = w0*p0[0] + w1*p1[0] + w2*p2[0];
  float py = w0*p0[1] + w1*p1[1] + w2*p2[1];
  float pz = w0*p0[2] + w1*p1[2] + w2*p2[2];
  smp[3*gid + 0] = px;
  smp[3*gid + 1] = py;
  smp[3*gid + 2] = pz;
  sqs[gid] = px*px + py*py + pz*pz;
}

__global__ void sqnorm_kernel(const float* __restrict__ pts, float* __restrict__ sq, int n) {
  int gid = blockIdx.x * blockDim.x + threadIdx.x;
  if (gid >= n) return;
  const float* p = pts + (size_t)3 * gid;
  sq[gid] = p[0]*p[0] + p[1]*p[1] + p[2]*p[2];
}

// ------------------------------------------------------------------
// Chamfer row-min via V_WMMA_F32_16X16X4_F32
// One wave per 16-row tile of X; sweeps all tiles of Y.
// A frag (16x4 f32): lanes 0-15 hold {K0,K1} for M=lane; lanes 16-31 hold {K2,K3}.
// B frag (4x16 f32): mirror with N=lane&15.
// D (16x16 f32): lane<16 -> (m=v, n=lane); lane>=16 -> (m=v+8, n=lane-16).
// Full tiles are branchless (clamp-free); the single partial tail tile uses
// clamped loads + inf penalty, so no exec-mask churn inside the hot loop.
// ------------------------------------------------------------------
__global__ __launch_bounds__(256) void chamfer_rowmin_kernel(
    const float* __restrict__ X,  const float* __restrict__ sqX,
    const float* __restrict__ Y,  const float* __restrict__ sqY,
    float* __restrict__ rowmin, int S, int T, int nB)
{
  const int wave = (blockIdx.x * blockDim.x + threadIdx.x) >> 5;
  const int lane = threadIdx.x & 31;
  const int tilesPerBatch = (S + 15) >> 4;
  if (wave >= tilesPerBatch * nB) return;            // wave-uniform exit
  const int b  = wave / tilesPerBatch;
  const int s0 = (wave - b * tilesPerBatch) << 4;

  const float* Xb   = X   + (size_t)b * S * 3;
  const float* Yb   = Y   + (size_t)b * T * 3;
  const float* sqXb = sqX + (size_t)b * S;
  const float* sqYb = sqY + (size_t)b * T;

  const int half = lane >> 4;   // 0: K0/K1, 1: K2/K3
  const int r    = lane & 15;   // M (for A) / N (for B) within tile

  // A fragment (resident across the whole Y sweep); clamp rows >= S to S-1
  // (their row-mins are computed but never written out).
  int m  = s0 + r;
  int mc = (m < S) ? m : (S - 1);
  const float* xp = Xb + 3 * mc;
  v2f a;
  a.x = xp[2 * half];            // K0 or K2
  a.y = half ? 0.0f : xp[1];     // K1 or pad (cndmask, both in-bounds)

  // |x|^2 for the 8 rows this lane's D elements belong to (clamped, unused if invalid)
  float nx[8];
  #pragma unroll
  for (int v = 0; v < 8; ++v) {
    int mm = s0 + (half << 3) + v;
    nx[v] = sqXb[(mm < S) ? mm : (S - 1)];
  }

  float rmin[8];
  #pragma unroll
  for (int v = 0; v < 8; ++v) rmin[v] = 3.0e38f;

  const int nFull = T >> 4;      // full 16-column tiles (no bounds checks)
  for (int tt = 0; tt < nFull; ++tt) {
    const int t = (tt << 4) + r;
    const float* yp = Yb + 3 * t;
    float c0 = yp[2 * half];
    float c1 = yp[1];
    v2f bf;
    bf.x = c0;
    bf.y = half ? 0.0f : c1;
    float ny = sqYb[t];

    v8f c = {};
    c = __builtin_amdgcn_wmma_f32_16x16x4_f32(
        /*neg_a=*/false, a, /*neg_b=*/false, bf,
        /*c_mod=*/(short)0, c, /*reuse_a=*/false, /*reuse_b=*/false);

    #pragma unroll
    for (int v = 0; v < 8; ++v) {
      float d = (nx[v] + ny) - 2.0f * c[v];
      rmin[v] = fminf(rmin[v], d);
    }
  }

  // tail tile (wave-uniform condition): clamped loads, +inf column penalty
  if ((nFull << 4) < T) {
    const int t  = (nFull << 4) + r;
    const int tc = (t < T) ? t : (T - 1);
    const float* yp = Yb + 3 * tc;
    float c0 = yp[2 * half];
    float c1 = yp[1];
    v2f bf;
    bf.x = c0;
    bf.y = half ? 0.0f : c1;
    float ny = (t < T) ? sqYb[t] : 3.0e38f;   // pad columns never win the min

    v8f c = {};
    c = __builtin_amdgcn_wmma_f32_16x16x4_f32(
        false, a, false, bf, (short)0, c, false, false);

    #pragma unroll
    for (int v = 0; v < 8; ++v) {
      float d = (nx[v] + ny) - 2.0f * c[v];
      rmin[v] = fminf(rmin[v], d);
    }
  }

  // reduce over the 16 columns: xor offsets 1,2,4,8 stay within each 16-lane half
  #pragma unroll
  for (int off = 8; off >= 1; off >>= 1) {
    #pragma unroll
    for (int v = 0; v < 8; ++v)
      rmin[v] = fminf(rmin[v], __shfl_xor(rmin[v], off, 32));
  }

  if (r == 0) {
    float* outp = rowmin + (size_t)b * S;
    #pragma unroll
    for (int v = 0; v < 8; ++v) {
      int mm = s0 + (half << 3) + v;
      if (mm < S) outp[mm] = rmin[v];
    }
  }
}

// ------------------------------------------------------------------
// simple atomic-sum reduction
// ------------------------------------------------------------------
__global__ void sum_kernel(const float* __restrict__ x, int n, float* __restrict__ acc) {
  int gid = blockIdx.x * blockDim.x + threadIdx.x;
  if (gid >= n) return;
  atomicAdd(acc, x[gid]);
}

// ------------------------------------------------------------------
// edge loss
// ------------------------------------------------------------------
__global__ void target_len_kernel(const float* __restrict__ vstd, const int* __restrict__ edges,
                                  float* __restrict__ acc, int E) {
  int gid = blockIdx.x * blockDim.x + threadIdx.x;
  if (gid >= E) return;
  int e0 = edges[2*gid + 0], e1 = edges[2*gid + 1];
  atomicAdd(acc, sqrtf(dist2p(vstd + 3*e0, vstd + 3*e1)));
}

__global__ void edge_loss_kernel(const float* __restrict__ verts_scr, const int* __restrict__ edges,
                                 const float* __restrict__ acc_tl, float* __restrict__ acc_edge,
                                 int E, int Vn, int nB) {
  int gid = blockIdx.x * blockDim.x + threadIdx.x;
  if (gid >= nB * E) return;
  int b = gid / E, e = gid - b * E;
  float tl = acc_tl[0] / (float)E;
  const float* vb = verts_scr + (size_t)b * Vn * 3;
  int e0 = edges[2*e + 0], e1 = edges[2*e + 1];
  float diff = sqrtf(dist2p(vb + 3*e0, vb + 3*e1)) - tl;
  atomicAdd(acc_edge, diff * diff);
}

// ------------------------------------------------------------------
// cotangent Laplacian scatter + finalize
// ------------------------------------------------------------------
__global__ void lap_scatter_kernel(const float* __restrict__ verts, const int* __restrict__ faces,
                                   float* __restrict__ Lv, float* __restrict__ rw,
                                   int F, int Vn, int nB) {
  int gid = blockIdx.x * blockDim.x + threadIdx.x;
  if (gid >= nB * F) return;
  int b = gid / F, f = gid - b * F;
  const float* vb = verts + (size_t)b * Vn * 3;
  int i0 = faces[3*f + 0], i1 = faces[3*f + 1], i2 = faces[3*f + 2];
  float cw[3];
  cot_weights(vb + 3*i0, vb + 3*i1, vb + 3*i2, cw);
  int sa[3] = {i1, i2, i0};
  int da[3] = {i2, i0, i1};
  #pragma unroll
  for (int k = 0; k < 3; ++k) {
    int s = sa[k], d = da[k];
    float wgt = cw[k];
    float* Ls = Lv + ((size_t)b * Vn + s) * 3;
    float* Ld = Lv + ((size_t)b * Vn + d) * 3;
    const float* ps = vb + 3*s;
    const float* pd = vb + 3*d;
    atomicAdd(&Ls[0], wgt * pd[0]); atomicAdd(&Ls[1], wgt * pd[1]); atomicAdd(&Ls[2], wgt * pd[2]);
    atomicAdd(&Ld[0], wgt * ps[0]); atomicAdd(&Ld[1], wgt * ps[1]); atomicAdd(&Ld[2], wgt * ps[2]);
    atomicAdd(&rw[(size_t)b * Vn + s], wgt);
    atomicAdd(&rw[(size_t)b * Vn + d], wgt);
  }
}

__global__ void lap_finalize_kernel(const float* __restrict__ verts_scr, const float* __restrict__ Lv,
                                    const float* __restrict__ rw, float* __restrict__ acc,
                                    int Vn, int nB) {
  int gid = blockIdx.x * blockDim.x + threadIdx.x;
  if (gid >= nB * Vn) return;
  float wsum = rw[gid];
  float inv = (wsum > 0.0f) ? 1.0f / fmaxf(wsum, 1e-12f) : 0.0f;
  const float* v = verts_scr + (size_t)3 * gid;
  float dx = Lv[3*gid + 0] * inv - v[0];
  float dy = Lv[3*gid + 1] * inv - v[1];
  float dz = Lv[3*gid + 2] * inv - v[2];
  atomicAdd(acc, sqrtf(dx*dx + dy*dy + dz*dz));
}

// ------------------------------------------------------------------
// ARAP rigid loss: scatter covariance, per-vertex Kabsch rotation, residual
// ------------------------------------------------------------------
__global__ void rigid_scatter_kernel(const float* __restrict__ vstd, const float* __restrict__ vscr,
                                     const int* __restrict__ faces, float* __restrict__ Smat,
                                     int F, int Vn, int nB) {
  int gid = blockIdx.x * blockDim.x + threadIdx.x;
  if (gid >= nB * F) return;
  int b = gid / F, f = gid - b * F;
  const float* sb = vscr + (size_t)b * Vn * 3;
  int i0 = faces[3*f + 0], i1 = faces[3*f + 1], i2 = faces[3*f + 2];
  float cw[3];
  cot_weights(vstd + 3*i0, vstd + 3*i1, vstd + 3*i2, cw);
  int sa[3] = {i1, i2, i0};
  int da[3] = {i2, i0, i1};
  #pragma unroll
  for (int k = 0; k < 3; ++k) {
    float wgt = cw[k];
    #pragma unroll
    for (int dir = 0; dir < 2; ++dir) {
      int s = dir ? da[k] : sa[k];
      int d = dir ? sa[k] : da[k];
      float es[3], ec[3];
      #pragma unroll
      for (int j = 0; j < 3; ++j) {
        es[j] = vstd[3*s + j] - vstd[3*d + j];
        ec[j] = sb[3*s + j]  - sb[3*d + j];
      }
      float* Sp = Smat + ((size_t)b * Vn + s) * 9;
      #pragma unroll
      for (int i = 0; i < 3; ++i)
        #pragma unroll
        for (int j = 0; j < 3; ++j)
          atomicAdd(&Sp[3*i + j], wgt * es[i] * ec[j]);
    }
  }
}

__device__ inline void jacobi_rotate(float A[3][3], float V[3][3], int p, int q) {
  float apq = A[p][q];
  if (fabsf(apq) < 1e-30f) return;
  float tau = (A[q][q] - A[p][p]) / (2.0f * apq);
  float t = (tau >= 0.0f ? 1.0f : -1.0f) / (fabsf(tau) + sqrtf(1.0f + tau*tau));
  float c = 1.0f / sqrtf(1.0f + t*t);
  float s = t * c;
  for (int k = 0; k < 3; ++k) {
    float akp = A[k][p], akq = A[k][q];
    A[k][p] = c*akp - s*akq;
    A[k][q] = s*akp + c*akq;
  }
  for (int k = 0; k < 3; ++k) {
    float apk = A[p][k], aqk = A[q][k];
    A[p][k] = c*apk - s*aqk;
    A[q][k] = s*apk + c*aqk;
  }
  for (int k = 0; k < 3; ++k) {
    float vkp = V[k][p], vkq = V[k][q];
    V[k][p] = c*vkp - s*vkq;
    V[k][q] = s*vkp + c*vkq;
  }
}

// in-place: Smat (3x3 covariance per vertex) -> R (row-major) per vertex
__global__ void rotation_kernel(float* __restrict__ Smat, int total) {
  int gid = blockIdx.x * blockDim.x + threadIdx.x;
  if (gid >= total) return;
  float* Sp = Smat + (size_t)gid * 9;
  float S[3][3];
  #pragma unroll
  for (int i = 0; i < 3; ++i)
    #pragma unroll
    for (int j = 0; j < 3; ++j) S[i][j] = Sp[3*i + j];

  float A[3][3];
  #pragma unroll
  for (int i = 0; i < 3; ++i)
    #pragma unroll
    for (int j = 0; j < 3; ++j)
      A[i][j] = S[0][i]*S[0][j] + S[1][i]*S[1][j] + S[2][i]*S[2][j];

  float V[3][3] = {{1,0,0},{0,1,0},{0,0,1}};
  for (int sw = 0; sw < 10; ++sw) {
    jacobi_rotate(A, V, 0, 1);
    jacobi_rotate(A, V, 0, 2);
    jacobi_rotate(A, V, 1, 2);
  }
  float ev[3] = {A[0][0], A[1][1], A[2][2]};
  int i0 = 0, i1 = 1, i2 = 2, tmp;
  if (ev[i0] < ev[i1]) { tmp = i0; i0 = i1; i1 = tmp; }
  if (ev[i0] < ev[i2]) { tmp = i0; i0 = i2; i2 = tmp; }
  if (ev[i1] < ev[i2]) { tmp = i1; i1 = i2; i2 = tmp; }

  float v0[3], v1[3], v2[3];
  #pragma unroll
  for (int k = 0; k < 3; ++k) { v0[k] = V[k][i0]; v1[k] = V[k][i1]; v2[k] = V[k][i2]; }

  // force det(V)=+1 so u2 = u0 x u1 yields the reference's det-fixed rotation
  float cx = v1[1]*v2[2] - v1[2]*v2[1];
  float cy = v1[2]*v2[0] - v1[0]*v2[2];
  float cz = v1[0]*v2[1] - v1[1]*v2[0];
  float det = v0[0]*cx + v0[1]*cy + v0[2]*cz;
  if (det < 0.0f) { v2[0] = -v2[0]; v2[1] = -v2[1]; v2[2] = -v2[2]; }

  float u0[3], u1[3], u2[3], R[3][3];
  #pragma unroll
  for (int i = 0; i < 3; ++i) u0[i] = S[i][0]*v0[0] + S[i][1]*v0[1] + S[i][2]*v0[2];
  float n0 = sqrtf(dot3p(u0, u0));
  if (n0 < 1e-18f) {
    R[0][0]=1; R[0][1]=0; R[0][2]=0;
    R[1][0]=0; R[1][1]=1; R[1][2]=0;
    R[2][0]=0; R[2][1]=0; R[2][2]=1;
  } else {
    float inv0 = 1.0f / n0;
    #pragma unroll
    for (int i = 0; i < 3; ++i) u0[i] *= inv0;
    #pragma unroll
    for (int i = 0; i < 3; ++i) u1[i] = S[i][0]*v1[0] + S[i][1]*v1[1] + S[i][2]*v1[2];
    float d01 = dot3p(u0, u1);
    #pragma unroll
    for (int i = 0; i < 3; ++i) u1[i] -= d01 * u0[i];
    float n1 = sqrtf(dot3p(u1, u1));
    if (n1 > 1e-12f * n0) {
      float inv1 = 1.0f / n1;
      #pragma unroll
      for (int i = 0; i < 3; ++i) u1[i] *= inv1;
    } else {
      float tvec[3];
      if (fabsf(u0[0]) < 0.9f) { tvec[0]=1; tvec[1]=0; tvec[2]=0; }
      else                     { tvec[0]=0; tvec[1]=1; tvec[2]=0; }
      float dt = dot3p(tvec, u0);
      #pragma unroll
      for (int i = 0; i < 3; ++i) tvec[i] -= dt * u0[i];
      float nt = sqrtf(dot3p(tvec, tvec));
      float invt = 1.0f / fmaxf(nt, 1e-20f);
      #pragma unroll
      for (int i = 0; i < 3; ++i) u1[i] = tvec[i] * invt;
    }
    u2[0] = u0[1]*u1[2] - u0[2]*u1[1];
    u2[1] = u0[2]*u1[0] - u0[0]*u1[2];
    u2[2] = u0[0]*u1[1] - u0[1]*u1[0];
    #pragma unroll
    for (int i = 0; i < 3; ++i)
      #pragma unroll
      for (int j = 0; j < 3; ++j)
        R[i][j] = v0[i]*u0[j] + v1[i]*u1[j] + v2[i]*u2[j];
  }
  #pragma unroll
  for (int i = 0; i < 3; ++i)
    #pragma unroll
    for (int j = 0; j < 3; ++j) Sp[3*i + j] = R[i][j];
}

__global__ void rigid_residual_kernel(const float* __restrict__ vstd, const float* __restrict__ vscr,
                                      const int* __restrict__ faces, const float* __restrict__ Rmat,
                                      float* __restrict__ acc, int F, int Vn, int nB) {
  int gid = blockIdx.x * blockDim.x + threadIdx.x;
  if (gid >= nB * F) return;
  int b = gid / F, f = gid - b * F;
  const float* sb = vscr + (size_t)b * Vn * 3;
  int i0 = faces[3*f + 0], i1 = faces[3*f + 1], i2 = faces[3*f + 2];
  float cw[3];
  cot_weights(vstd + 3*i0, vstd + 3*i1, vstd + 3*i2, cw);
  int sa[3] = {i1, i2, i0};
  int da[3] = {i2, i0, i1};
  float sum = 0.0f;
  #pragma unroll
  for (int k = 0; k < 3; ++k) {
    float wgt = cw[k];
    #pragma unroll
    for (int dir = 0; dir < 2; ++dir) {
      int s = dir ? da[k] : sa[k];
      int d = dir ? sa[k] : da[k];
      float es[3], ec[3];
      #pragma unroll
      for (int j = 0; j < 3; ++j) {
        es[j] = vstd[3*s + j] - vstd[3*d + j];
        ec[j] = sb[3*s + j]  - sb[3*d + j];
      }
      const float* R = Rmat + ((size_t)b * Vn + s) * 9;
      #pragma unroll
      for (int i = 0; i < 3; ++i) {
        float ri = ec[i] - (R[3*i + 0]*es[0] + R[3*i + 1]*es[1] + R[3*i + 2]*es[2]);
        sum += wgt * ri * ri;
      }
    }
  }
  atomicAdd(acc, sum);
}

// ------------------------------------------------------------------
// finalize 5 outputs
// ------------------------------------------------------------------
__global__ void finalize_kernel(const float* __restrict__ acc, float* __restrict__ out,
                                int S, int T, int E, int Vn, int nB) {
  if (blockIdx.x == 0 && threadIdx.x == 0) {
    out[0] = acc[0] / ((float)nB * (float)S) + acc[1] / ((float)nB * (float)T);
    out[1] = 1e-5f;
    out[2] = acc[4] / ((float)nB * (float)Vn);
    out[3] = acc[3] / ((float)nB * (float)E);
    out[4] = acc[5] / ((float)nB * (float)Vn);
  }
}

// ------------------------------------------------------------------
// host launcher
// ------------------------------------------------------------------
extern "C" void kernel_launch(void* const* d_in, const int* in_sizes, int n_in,
                              void* d_out, int out_size, void* d_ws, size_t ws_size,
                              hipStream_t stream) {
  (void)n_in; (void)out_size; (void)ws_size;
  const float* verts_scr = (const float*)d_in[0];
  const float* verts_std = (const float*)d_in[1];
  const float* trg       = (const float*)d_in[2];
  const float* u_face    = (const float*)d_in[3];
  const float* r_bary    = (const float*)d_in[4];
  const int*   faces     = (const int*)d_in[5];
  const int*   edges     = (const int*)d_in[6];
  float* out = (float*)d_out;

  const int Vn = in_sizes[1] / 3;
  const int nB = in_sizes[0] / in_sizes[1];
  const int S  = in_sizes[2] / (3 * nB);
  const int T  = S;
  const int F  = in_sizes[5] / 3;
  const int E  = in_sizes[6] / 2;
  const int nblk = (F + 255) / 256;

  // workspace layout (floats)
  float* w = (float*)d_ws;
  size_t o = 0;
  float* acc   = w + o; o += 8;
  float* area  = w + o; o += (size_t)nB * F;
  float* pref  = w + o; o += (size_t)nB * F;
  float* bsum  = w + o; o += (size_t)nB * nblk;
  float* smp   = w + o; o += (size_t)nB * S * 3;
  float* sqs   = w + o; o += (size_t)nB * S;
  float* sqt   = w + o; o += (size_t)nB * T;
  float* minA  = w + o; o += (size_t)nB * S;
  float* minB  = w + o; o += (size_t)nB * T;
  float* Lv    = w + o; o += (size_t)nB * Vn * 3;   // Lv, rw, Smat contiguous (zeroed together)
  float* rw    = w + o; o += (size_t)nB * Vn;
  float* Smat  = w + o; o += (size_t)nB * Vn * 9;

  auto g = [](int n) { return (n + THREADS - 1) / THREADS; };

  // zero accumulators + scatter targets
  zero_kernel<<<1, 32, 0, stream>>>(acc, 8);
  zero_kernel<<<g(nB * Vn * 13), THREADS, 0, stream>>>(Lv, nB * Vn * 13);

  // --- sampling: areas -> inclusive CDF -> sample points ---
  face_area_kernel<<<g(nB * F), THREADS, 0, stream>>>(verts_scr, faces, area, F, Vn, nB);
  dim3 sg(nblk, nB);
  scan_block_kernel<<<sg, 256, 0, stream>>>(area, pref, bsum, F, nblk);
  scan_partials_kernel<<<1, 32, 0, stream>>>(bsum, nblk, nB);
  scan_add_kernel<<<sg, 256, 0, stream>>>(pref, bsum, F, nblk);
  sample_kernel<<<g(nB * S), THREADS, 0, stream>>>(verts_scr, faces, pref, u_face, r_bary,
                                                   smp, sqs, F, Vn, S, nB);
  sqnorm_kernel<<<g(nB * T), THREADS, 0, stream>>>(trg, sqt, nB * T);

  // --- chamfer via WMMA (two symmetric passes) ---
  int tilesA = nB * ((S + 15) / 16);
  int tilesB = nB * ((T + 15) / 16);
  chamfer_rowmin_kernel<<<(tilesA + 7) / 8, 256, 0, stream>>>(smp, sqs, trg, sqt, minA, S, T, nB);
  chamfer_rowmin_kernel<<<(tilesB + 7) / 8, 256, 0, stream>>>(trg, sqt, smp, sqs, minB, T, S, nB);
  sum_kernel<<<g(nB * S), THREADS, 0, stream>>>(minA, nB * S, acc + 0);
  sum_kernel<<<g(nB * T), THREADS, 0, stream>>>(minB, nB * T, acc + 1);

  // --- edge loss ---
  target_len_kernel<<<g(E), THREADS, 0, stream>>>(verts_std, edges, acc + 2, E);
  edge_loss_kernel<<<g(nB * E), THREADS, 0, stream>>>(verts_scr, edges, acc + 2, acc + 3, E, Vn, nB);

  // --- Laplacian loss ---
  lap_scatter_kernel<<<g(nB * F), THREADS, 0, stream>>>(verts_scr, faces, Lv, rw, F, Vn, nB);
  lap_finalize_kernel<<<g(nB * Vn), THREADS, 0, stream>>>(verts_scr, Lv, rw, acc + 4, Vn, nB);

  // --- rigid (ARAP) loss ---
  rigid_scatter_kernel<<<g(nB * F), THREADS, 0, stream>>>(verts_std, verts_scr, faces, Smat, F, Vn, nB);
  rotation_kernel<<<g(nB * Vn), THREADS, 0, stream>>>(Smat, nB * Vn);
  rigid_residual_kernel<<<g(nB * F), THREADS, 0, stream>>>(verts_std, verts_scr, faces, Smat,
                                                           acc + 5, F, Vn, nB);

  finalize_kernel<<<1, 32, 0, stream>>>(acc, out, S, T, E, Vn, nB);
}